// FractalLoss_40690520163157
// MI455X (gfx1250) — compile-verified
//
#include <hip/hip_runtime.h>

typedef float v2f __attribute__((ext_vector_type(2)));
typedef float v8f __attribute__((ext_vector_type(8)));

#define IMG_H 1024
#define IMG_W 1024
#define NBATCH 16
#define NCH 3

// counts layout: counts[s*10 + img*5 + level], img 0=fake 1=target, level 0..4 (box 2..32)
__global__ void zero_counts(unsigned int* __restrict__ counts) {
    int t = threadIdx.x;
    if (t < NBATCH * 2 * 5) counts[t] = 0u;
}

__global__ __launch_bounds__(256) void fractal_boxcount(
    const float* __restrict__ fake,
    const float* __restrict__ target,
    const float* __restrict__ mask,
    unsigned int* __restrict__ counts)
{
    __shared__ unsigned int lf[32 * 33];   // stride 33: avoid 32-way bank conflicts
    __shared__ unsigned int lt[32 * 33];
    __shared__ unsigned int wg[10];

    const int tid = threadIdx.x;
    const int sy  = blockIdx.x;   // 0..31: strip of 32 rows
    const int b   = blockIdx.y;   // 0..15: sample

    if (tid < 10) wg[tid] = 0u;

    const int wave = tid >> 5;
    const int lane = tid & 31;

    const size_t hw    = (size_t)IMG_H * IMG_W;
    const size_t fbase = (size_t)b * NCH * hw;
    const size_t mbase = (size_t)b * hw;
    const int y0 = sy * 32;

    // Each wave produces 32-bit occupancy words via ballot: word (ly, tx)
    // covers pixels [y0+ly][32*tx .. 32*tx+31]. 1024 words per strip, 8 waves.
    for (int idx = wave; idx < 1024; idx += 8) {
        const int ly = idx >> 5;
        const int tx = idx & 31;
        const size_t p = (size_t)(y0 + ly) * IMG_W + (size_t)(tx * 32 + lane);

        float f0 = __builtin_nontemporal_load(fake   + fbase + p);
        float f1 = __builtin_nontemporal_load(fake   + fbase + hw + p);
        float f2 = __builtin_nontemporal_load(fake   + fbase + 2 * hw + p);
        float t0 = __builtin_nontemporal_load(target + fbase + p);
        float t1 = __builtin_nontemporal_load(target + fbase + hw + p);
        float t2 = __builtin_nontemporal_load(target + fbase + 2 * hw + p);
        float mm = __builtin_nontemporal_load(mask   + mbase + p);

        float fr = (f0 + f1 + f2) * (1.0f / 3.0f) * mm;
        float tr = (t0 + t1 + t2) * (1.0f / 3.0f) * mm;

        unsigned int bf = __builtin_amdgcn_ballot_w32(fr > 0.5f);
        unsigned int bt = __builtin_amdgcn_ballot_w32(tr > 0.5f);
        if (lane == 0) { lf[ly * 33 + tx] = bf; lt[ly * 33 + tx] = bt; }
    }
    __syncthreads();

    // One thread per 32x32 tile (32 fake tiles + 32 target tiles).
    if (tid < 64) {
        const int tx = tid & 31;
        const unsigned int* __restrict__ src = (tid < 32) ? lf : lt;
        unsigned int r2[16], r4[8], r8[4], r16[2];
        unsigned int c1 = 0, c2 = 0, c3 = 0, c4 = 0, c5 = 0;

        #pragma unroll
        for (int i = 0; i < 16; ++i) {            // box size 2
            unsigned int r = src[(2 * i) * 33 + tx] | src[(2 * i + 1) * 33 + tx];
            r2[i] = r;
            unsigned int bb = (r | (r >> 1)) & 0x55555555u;
            c1 += (unsigned int)__popc(bb);
        }
        #pragma unroll
        for (int i = 0; i < 8; ++i) {             // box size 4
            unsigned int r = r2[2 * i] | r2[2 * i + 1];
            r4[i] = r;
            unsigned int bb = r | (r >> 1); bb |= bb >> 2; bb &= 0x11111111u;
            c2 += (unsigned int)__popc(bb);
        }
        #pragma unroll
        for (int i = 0; i < 4; ++i) {             // box size 8
            unsigned int r = r4[2 * i] | r4[2 * i + 1];
            r8[i] = r;
            unsigned int bb = r | (r >> 1); bb |= bb >> 2; bb |= bb >> 4; bb &= 0x01010101u;
            c3 += (unsigned int)__popc(bb);
        }
        #pragma unroll
        for (int i = 0; i < 2; ++i) {             // box size 16
            unsigned int r = r8[2 * i] | r8[2 * i + 1];
            r16[i] = r;
            unsigned int bb = r | (r >> 1); bb |= bb >> 2; bb |= bb >> 4; bb |= bb >> 8;
            bb &= 0x00010001u;
            c4 += (unsigned int)__popc(bb);
        }
        unsigned int r32 = r16[0] | r16[1];       // box size 32
        c5 = r32 ? 1u : 0u;

        const int base = (tid < 32) ? 0 : 5;
        atomicAdd(&wg[base + 0], c1);
        atomicAdd(&wg[base + 1], c2);
        atomicAdd(&wg[base + 2], c3);
        atomicAdd(&wg[base + 3], c4);
        atomicAdd(&wg[base + 4], c5);
    }
    __syncthreads();
    if (tid < 10) atomicAdd(&counts[b * 10 + tid], wg[tid]);
}

// Final fit: fd = -slope of log(counts) vs log(sizes). With sizes 2^1..2^5,
// weight for level i is (i-2)/(10 ln2); the i=2 weight is 0, so the fit is a
// K=4 dot product -> V_WMMA_F32_16X16X4_F32 with A = delta-log-counts (M=sample),
// B = broadcast weights (-2,-1,1,2). Launch with exactly one full wave (EXEC all 1s).
__global__ void fractal_fd_final(const unsigned int* __restrict__ counts,
                                 float* __restrict__ out)
{
    const int lane = threadIdx.x;        // 0..31
    const int m  = lane & 15;            // sample row of A
    const int kh = lane >> 4;            // 0: K={0,1} (levels 0,1); 1: K={2,3} (levels 3,4)

    const int   l0 = kh ? 3 : 0;
    const int   l1 = kh ? 4 : 1;
    const float w0 = kh ? 1.0f : -2.0f;
    const float w1 = kh ? 2.0f : -1.0f;

    const float cf0 = (float)counts[m * 10 + 0 * 5 + l0];
    const float ct0 = (float)counts[m * 10 + 1 * 5 + l0];
    const float cf1 = (float)counts[m * 10 + 0 * 5 + l1];
    const float ct1 = (float)counts[m * 10 + 1 * 5 + l1];

    v2f a, bw;
    a.x = logf(cf0 + 1e-8f) - logf(ct0 + 1e-8f);
    a.y = logf(cf1 + 1e-8f) - logf(ct1 + 1e-8f);
    bw.x = w0;
    bw.y = w1;

    v8f c = {};
    v8f d = __builtin_amdgcn_wmma_f32_16x16x4_f32(
        /*neg_a=*/false, a, /*neg_b=*/false, bw,
        /*c_mod=*/(short)0, c, /*reuse_a=*/false, /*reuse_b=*/false);

    // C/D layout: VGPR r, lanes 0-15 -> M=r (N=lane); lanes 16-31 -> M=8+r.
    // B is column-broadcast, so every N column holds the same dot product;
    // lane 0 has rows 0..7, lane 16 has rows 8..15.
    float s = 0.0f;
    #pragma unroll
    for (int i = 0; i < 8; ++i) s += fabsf(d[i]);

    float other = __shfl(s, lane ^ 16, 32);
    if (lane == 0) {
        const float scale = 1.0f / (10.0f * 0.6931471805599453f * 16.0f);
        out[0] = (s + other) * scale;
    }
}

extern "C" void kernel_launch(void* const* d_in, const int* in_sizes, int n_in,
                              void* d_out, int out_size, void* d_ws, size_t ws_size,
                              hipStream_t stream) {
    const float* fake   = (const float*)d_in[0];
    const float* target = (const float*)d_in[1];
    const float* mask   = (const float*)d_in[2];
    unsigned int* counts = (unsigned int*)d_ws;
    float* out = (float*)d_out;

    zero_counts<<<1, 256, 0, stream>>>(counts);
    dim3 grid(32, NBATCH);                         // 32 row-strips x 16 samples
    fractal_boxcount<<<grid, 256, 0, stream>>>(fake, target, mask, counts);
    fractal_fd_final<<<1, 32, 0, stream>>>(counts, out);
}